// GCN_40029095198810
// MI455X (gfx1250) — compile-verified
//
#include <hip/hip_runtime.h>
#include <hip/hip_bf16.h>

// GCN, 4 layers:  h = A_norm @ (X W) + b  per layer, then reshape+pad.
//   1) gemm_wmma<NT>: H = X @ W  (fp32 WMMA 16x16x4; W staged zero-padded &
//      k-pair-interleaved in LDS so each B fragment is a single ds_load_b64;
//      one wave computes a 16-row strip across all NT column tiles, A-fragment
//      reuse, batched B loads -> branch-free wmma-dense inner loop)
//   2) selfloop_bias: Y[n] = H[n]*dinv[n]^2 + b   (writes full padded row)
//   3) edge_agg:  Y[dst] += H[src]*dinv[src]*dinv[dst]  (fp32 atomics, wave/edge)
// Final pack_pad writes the (B,45,128) padded output with float4 stores.

typedef __attribute__((ext_vector_type(2))) float v2f;
typedef __attribute__((ext_vector_type(4))) float v4f;
typedef __attribute__((ext_vector_type(8))) float v8f;

#define D_IN   79
#define KPAD1  80    // layer-1 K padded to multiple of 4
#define GS     32
#define PADR   45
#define DOUT   128
#define FMAX   144   // row stride of activation buffers (>=138, mult of 16)

// ---------------------------------------------------------------- degree prep
__global__ void init_deg(float* __restrict__ deg, int n) {
    int i = blockIdx.x * blockDim.x + threadIdx.x;
    if (i < n) deg[i] = 1.0f;                       // self-loop contributes 1
}

__global__ void accum_deg(const int* __restrict__ dst, float* __restrict__ deg, int e) {
    int i = blockIdx.x * blockDim.x + threadIdx.x;
    if (i < e) atomicAdd(&deg[dst[i]], 1.0f);
}

__global__ void finish_dinv(float* __restrict__ deg, int n) {
    int i = blockIdx.x * blockDim.x + threadIdx.x;
    if (i < n) deg[i] = rsqrtf(deg[i]);             // deg >= 1 always
}

// ---------------------------------------------------------------- pad x to stride 80
__global__ void pad_x(const float* __restrict__ x, float* __restrict__ xp, long long total) {
    long long i = (long long)blockIdx.x * blockDim.x + threadIdx.x;
    if (i >= total) return;
    int k = (int)(i % KPAD1);
    long long n = i / KPAD1;
    xp[i] = (k < D_IN) ? x[n * D_IN + k] : 0.0f;
}

// ---------------------------------------------------------------- fp32 WMMA GEMM
// H[M x NT*16] = X[M x Kpad] * Wpad[Kpad x NT*16].
// LDS layout: k-pair interleaved: lds[((k>>1)*NTF + c)*2 + (k&1)] = Wpad[k][c]
//   -> B fragment {W[ka][col], W[ka+1][col]} = one aligned ds_load_b64.
// A frag (16x4): lane l: row m0+(l&15); VGPRs = X[row][k0+2*(l>>4) .. +1] (b64)
// C/D (16x16):   VGPR r: row m0+r+8*(l>>4), col (l&15) of tile
template <int NT>
__global__ void gemm_wmma(const float* __restrict__ X, int ldx, int Kpad,
                          const float* __restrict__ W, int ldw, int K, int Fout,
                          float* __restrict__ H, int ldh, int Mtiles) {
    extern __shared__ float ldsW[];                 // [Kpad/2][NTF][2], zero-padded
    const int NTF = NT * 16;

    for (int i = threadIdx.x; i < Kpad * NTF; i += blockDim.x) {
        int k = i / NTF;
        int c = i - k * NTF;
        float v = (k < K && c < Fout) ? W[k * ldw + c] : 0.0f;
        ldsW[(((k >> 1) * NTF) + c) * 2 + (k & 1)] = v;
    }
    __syncthreads();

    const int lane = threadIdx.x & 31;
    const int mt   = blockIdx.x * (blockDim.x >> 5) + (threadIdx.x >> 5);
    if (mt >= Mtiles) return;

    const int half = lane >> 4;                     // 0: lanes 0-15, 1: 16-31
    const int l16  = lane & 15;
    const int m0   = mt << 4;
    const float* xrow = X + (size_t)(m0 + l16) * ldx;
    const v2f*   wp0 = (const v2f*)ldsW;

    v8f acc[NT];
#pragma unroll
    for (int t = 0; t < NT; ++t) acc[t] = v8f{};

    for (int k0 = 0; k0 < Kpad; k0 += 4) {
        const int ka = k0 + (half << 1);
        v2f a = *(const v2f*)(xrow + ka);           // global_load_b64

        const v2f* wp = wp0 + ((k0 >> 1) + half) * NTF + l16;
        v2f b[NT];
#pragma unroll
        for (int t = 0; t < NT; ++t) b[t] = wp[t * 16];   // ds_load_b64 each

#pragma unroll
        for (int t = 0; t < NT; ++t)
            // (neg_a, A, neg_b, B, c_mod, C, reuse_a, reuse_b)
            acc[t] = __builtin_amdgcn_wmma_f32_16x16x4_f32(false, a, false, b[t],
                                                           (short)0, acc[t],
                                                           false, false);
    }

#pragma unroll
    for (int t = 0; t < NT; ++t) {
        const int col = t * 16 + l16;
#pragma unroll
        for (int r = 0; r < 8; ++r)
            H[(size_t)(m0 + r + (half << 3)) * ldh + col] = acc[t][r];
    }
}

// ---------------------------------------------------------------- aggregation
// Writes the full FMAX-wide row: pad columns of H are exact zeros (zero-padded
// W), pad bias is 0, so Y pad columns end up 0 -> next layer reads Kpad=144
// without bounds checks.
__global__ void selfloop_bias(const float* __restrict__ H, const float* __restrict__ dinv,
                              const float* __restrict__ bias, float* __restrict__ Y,
                              int Fout, long long total) {
    long long i = (long long)blockIdx.x * blockDim.x + threadIdx.x;
    if (i >= total) return;
    int f = (int)(i % FMAX);
    int n = (int)(i / FMAX);
    float d = dinv[n];
    float b = (f < Fout) ? bias[f] : 0.0f;
    Y[i] = H[i] * d * d + b;
}

__global__ void edge_agg(const float* __restrict__ H, const float* __restrict__ dinv,
                         const int* __restrict__ src, const int* __restrict__ dst,
                         float* __restrict__ Y, int Fout, int E) {
    int e    = (blockIdx.x * blockDim.x + threadIdx.x) >> 5;
    int lane = threadIdx.x & 31;
    if (e >= E) return;
    int s = src[e];
    int d = dst[e];
    float norm = dinv[s] * dinv[d];
    const float* hs = H + (size_t)s * FMAX;
    float*       yd = Y + (size_t)d * FMAX;
    for (int f = lane; f < Fout; f += 32)
        atomicAdd(&yd[f], hs[f] * norm);
}

// ---------------------------------------------------------------- final pack+pad
// float4-vectorized: DOUT=128 -> 32 vec4 per row; FMAX row stride is 16B-aligned.
__global__ void pack_pad(const float* __restrict__ Y, float* __restrict__ out,
                         long long total4) {
    long long i = (long long)blockIdx.x * blockDim.x + threadIdx.x;
    if (i >= total4) return;
    const int V = DOUT / 4;                         // 32 vec4 per padded row
    int       f = (int)(i % V);
    long long t = i / V;
    int       g = (int)(t % PADR);
    long long b = t / PADR;
    v4f v = {};
    if (g < GS) v = *(const v4f*)(Y + (b * GS + g) * (long long)FMAX + f * 4);
    *((v4f*)out + i) = v;
}

// ---------------------------------------------------------------- launch
extern "C" void kernel_launch(void* const* d_in, const int* in_sizes, int n_in,
                              void* d_out, int out_size, void* d_ws, size_t ws_size,
                              hipStream_t stream) {
    const float* x  = (const float*)d_in[0];
    const int*   ei = (const int*)d_in[1];
    const float* W[4]  = { (const float*)d_in[2], (const float*)d_in[4],
                           (const float*)d_in[6], (const float*)d_in[8] };
    const float* Bv[4] = { (const float*)d_in[3], (const float*)d_in[5],
                           (const float*)d_in[7], (const float*)d_in[9] };
    const int Kd[4]    = { D_IN, 138, 138, 138 };
    const int Kpad[4]  = { KPAD1, FMAX, FMAX, FMAX };
    const int FoutA[4] = { 138, 138, 138, DOUT };

    const int N = in_sizes[0] / D_IN;
    const int E = in_sizes[1] / 2;
    const int* src = ei;
    const int* dst = ei + E;

    float* dinv = (float*)d_ws;
    float* bufA = dinv + N;                    // GEMM output H  [N x FMAX]
    float* bufB = bufA + (size_t)N * FMAX;     // layer output Y [N x FMAX] (also padded X)

    const int TB = 256;

    // degree -> dinv (in place)
    init_deg   <<<(N + TB - 1) / TB, TB, 0, stream>>>(dinv, N);
    accum_deg  <<<(E + TB - 1) / TB, TB, 0, stream>>>(dst, dinv, E);
    finish_dinv<<<(N + TB - 1) / TB, TB, 0, stream>>>(dinv, N);

    // pad x -> bufB (stride 80); consumed by layer-1 GEMM before bufB is reused
    {
        long long total = (long long)N * KPAD1;
        pad_x<<<(unsigned)((total + TB - 1) / TB), TB, 0, stream>>>(x, bufB, total);
    }

    const int Mtiles  = (N + 15) / 16;
    const int gblocks = (Mtiles + (TB / 32) - 1) / (TB / 32);

    const float* Xin = bufB;
    int ldx = KPAD1;
    for (int l = 0; l < 4; ++l) {
        const int K = Kd[l], Kp = Kpad[l], Fo = FoutA[l];
        const int NT = (Fo + 15) / 16;
        const size_t shmem = (size_t)Kp * NT * 16 * sizeof(float);

        if (NT == 9)
            gemm_wmma<9><<<gblocks, TB, shmem, stream>>>(Xin, ldx, Kp, W[l], Fo, K, Fo,
                                                         bufA, FMAX, Mtiles);
        else
            gemm_wmma<8><<<gblocks, TB, shmem, stream>>>(Xin, ldx, Kp, W[l], Fo, K, Fo,
                                                         bufA, FMAX, Mtiles);

        {
            long long total = (long long)N * FMAX;
            selfloop_bias<<<(unsigned)((total + TB - 1) / TB), TB, 0, stream>>>(
                bufA, dinv, Bv[l], bufB, Fo, total);
        }
        {
            long long threads = (long long)E * 32;
            edge_agg<<<(unsigned)((threads + TB - 1) / TB), TB, 0, stream>>>(
                bufA, dinv, src, dst, bufB, Fo, E);
        }
        Xin = bufB;
        ldx = FMAX;
    }

    // (B, 45, 128) padded output (vec4)
    {
        long long total4 = (long long)out_size / 4;
        pack_pad<<<(unsigned)((total4 + TB - 1) / TB), TB, 0, stream>>>(
            bufB, (float*)d_out, total4);
    }
}